// ResiduePose_29231547417075
// MI455X (gfx1250) — compile-verified
//
#include <hip/hip_runtime.h>

// ---------------------------------------------------------------------------
// ResiduePose: coords[b,l,a,i] = sum_j R[b,l,i,j] * P[a,j] + (t[b,l,i] - mean_b t)
//   B=32, L=65536, 4 atoms, 3 coords.  Memory-bound (~160MB @ 23.3TB/s ~ 7us).
//   Rotation apply via V_WMMA_F32_16X16X4_F32, 16 residues per wave-tile.
//   Translation is folded into the matmul's unused K=3 slot:
//     A(l,0..2) = R_l(i,:),  A(l,3) = t_i(l),  B(3,a) = 1  =>  D = R*P^T + t
//   so C = 0 and no cross-lane broadcast is needed at all.
//   Output staged through LDS so global stores are fully coalesced b128.
// ---------------------------------------------------------------------------

typedef __attribute__((ext_vector_type(2))) float v2f;
typedef __attribute__((ext_vector_type(8))) float v8f;

__device__ __constant__ float P_IC[4][3] = {
    { 1.460091f,    0.0f,         0.0f        },   // N
    { 0.0f,         0.0f,         0.0f        },   // CA
    {-0.56431316f,  1.41695817f,  0.0f        },   // C
    {-0.52426314f, -0.76611338f,  1.20561194f },   // CB
};

// ---------------- pass 1: per-batch sum of translation --------------------
__global__ void __launch_bounds__(256)
mean_reduce_kernel(const float* __restrict__ trans, float* __restrict__ sums, int L)
{
    const int b   = blockIdx.y;
    const int tid = threadIdx.x;
    const int start = blockIdx.x * (256 * 16);
    const size_t base = (size_t)b * (size_t)L;

    float sx = 0.f, sy = 0.f, sz = 0.f;
    #pragma unroll 4
    for (int k = 0; k < 16; ++k) {
        int idx = start + k * 256 + tid;
        if (idx < L) {
            size_t t3 = (base + (size_t)idx) * 3;
            sx += trans[t3 + 0];
            sy += trans[t3 + 1];
            sz += trans[t3 + 2];
        }
    }

    __shared__ float red[3][256];
    red[0][tid] = sx; red[1][tid] = sy; red[2][tid] = sz;
    __syncthreads();
    for (int s = 128; s > 0; s >>= 1) {
        if (tid < s) {
            red[0][tid] += red[0][tid + s];
            red[1][tid] += red[1][tid + s];
            red[2][tid] += red[2][tid + s];
        }
        __syncthreads();
    }
    if (tid == 0) {
        atomicAdd(&sums[b * 3 + 0], red[0][0]);   // global_atomic_add_f32
        atomicAdd(&sums[b * 3 + 1], red[1][0]);
        atomicAdd(&sums[b * 3 + 2], red[2][0]);
    }
}

// ---------------- pass 2: WMMA transform, 16 residues per wave-tile -------
__global__ void __launch_bounds__(256)
residue_pose_kernel(const float*  __restrict__ trans,
                    const float4* __restrict__ quat,
                    const float*  __restrict__ sums,
                    float*        __restrict__ out,
                    int L, int nTiles)
{
    __shared__ __align__(16) float lds[8][192];   // 16 residues * 12 floats per wave

    const int  lane  = threadIdx.x & 31;
    const int  wslot = threadIdx.x >> 5;
    const int  wid   = blockIdx.x * (blockDim.x >> 5) + wslot;
    const int  nWav  = gridDim.x * (blockDim.x >> 5);
    const bool hi    = lane >= 16;
    const int  c     = lane & 15;
    const float invL = 1.0f / (float)L;

    // B operand: B(K=j, col=a) = P[a][j] for j<3; B(3, a) = 1 (translation row).
    // Cols 4..15 zero. Layout: v0 = rows K0|K2, v1 = rows K1|K3.
    v2f Bv; Bv[0] = 0.f; Bv[1] = 0.f;
    if (c < 4) {
        Bv[0] = hi ? P_IC[c][2] : P_IC[c][0];
        Bv[1] = hi ? 1.0f       : P_IC[c][1];
    }

    for (int tile = wid; tile < nTiles; tile += nWav) {
        const size_t res0 = (size_t)tile << 4;                 // first residue of tile
        const size_t r    = res0 + (size_t)c;                  // this lane's residue
        const int    b    = __builtin_amdgcn_readfirstlane((int)(res0 / (size_t)L));

        // per-lane residue data (lanes 16..31 mirror lanes 0..15; loads merge).
        // The mirror is what lets lane 16+m supply A-row m's K=2/K=3 elements.
        float4 q = quat[r];                                    // global_load_b128
        size_t t3 = r * 3;
        float tx = trans[t3], ty = trans[t3 + 1], tz = trans[t3 + 2];

        // uniform address -> scalar loads via constant cache
        float mx = sums[b * 3 + 0] * invL;
        float my = sums[b * 3 + 1] * invL;
        float mz = sums[b * 3 + 2] * invL;
        float tcv[3] = { tx - mx, ty - my, tz - mz };

        float nrm = sqrtf(q.x * q.x + q.y * q.y + q.z * q.z + q.w * q.w);
        float s   = 1.0f / (nrm + 1e-6f);
        float qr = q.x * s, qi = q.y * s, qj = q.z * s, qk = q.w * s;

        float R[3][3];
        R[0][0] = 1.f - 2.f * (qj * qj + qk * qk);
        R[0][1] = 2.f * (qi * qj - qk * qr);
        R[0][2] = 2.f * (qi * qk + qj * qr);
        R[1][0] = 2.f * (qi * qj + qk * qr);
        R[1][1] = 1.f - 2.f * (qi * qi + qk * qk);
        R[1][2] = 2.f * (qj * qk - qi * qr);
        R[2][0] = 2.f * (qi * qk - qj * qr);
        R[2][1] = 2.f * (qj * qk + qi * qr);
        R[2][2] = 1.f - 2.f * (qi * qi + qj * qj);

        // One WMMA per output coordinate i:
        //   D(l,a) = sum_{j<3} R_l(i,j) P(a,j) + t_i(l) * B(3,a)=1,  C = 0.
        // A layout: lanes 0-15 -> {K0,K1}, lanes 16-31 -> {K2,K3}.
        #pragma unroll
        for (int i = 0; i < 3; ++i) {
            v2f Av;
            Av[0] = hi ? R[i][2] : R[i][0];
            Av[1] = hi ? tcv[i]  : R[i][1];

            v8f Cz = {};                                       // inline-0 accumulator
            v8f D = __builtin_amdgcn_wmma_f32_16x16x4_f32(
                false, Av, false, Bv, (short)0, Cz, false, false);

            // D: vgpr v, lane n -> (l = v + (n>=16 ? 8:0), a = n&15); useful a<4
            if (c < 4) {
                #pragma unroll
                for (int v = 0; v < 8; ++v)
                    lds[wslot][(v + (hi ? 8 : 0)) * 12 + c * 3 + i] = D[v];
            }
        }

        // intra-wave LDS write->read ordering (same-wave DS ops are in-order;
        // just need completion before the b128 readback)
        asm volatile("s_wait_dscnt 0" ::: "memory");

        // coalesced b128 stores: 48 float4 = 192 floats = 16 residues * 12
        const float4* lsrc = (const float4*)&lds[wslot][0];
        float4* gdst = (float4*)(out + res0 * 12);
        gdst[lane] = lsrc[lane];
        if (lane < 16) gdst[32 + lane] = lsrc[32 + lane];
    }
}

// ---------------------------------------------------------------------------
extern "C" void kernel_launch(void* const* d_in, const int* in_sizes, int n_in,
                              void* d_out, int out_size, void* d_ws, size_t ws_size,
                              hipStream_t stream)
{
    const float*  trans = (const float*)d_in[0];   // (B,L,3) f32
    const float4* quat  = (const float4*)d_in[1];  // (B,L,4) f32
    float* out = (float*)d_out;                    // (B,L,4,3) f32

    const int B = 32;
    size_t nRes = (size_t)in_sizes[0] / 3;         // B*L
    int L = (int)(nRes / (size_t)B);

    float* sums = (float*)d_ws;                    // 32*3 f32 partial sums
    hipMemsetAsync(d_ws, 0, (size_t)B * 3 * sizeof(float), stream);

    int gx = (L + 4095) / 4096;                    // 256 thr * 16 elems per block
    mean_reduce_kernel<<<dim3(gx, B), 256, 0, stream>>>(trans, sums, L);

    int nTiles = (int)(nRes / 16);
    residue_pose_kernel<<<512, 256, 0, stream>>>(trans, quat, sums, out, L, nTiles);
}